// GNNDecoder_46978352284504
// MI455X (gfx1250) — compile-verified
//
#include <hip/hip_runtime.h>

// GNN decoder pipeline for MI455X (gfx1250, wave32).
// GEMMs run on the matrix pipe via V_WMMA_F32_16X16X4_F32 (fp32-exact).
// Edge aggregation is HBM-bandwidth bound -> one wave32 per edge,
// float4/lane gather + global_atomic_add_f32 scatter (L2-resident dst).

typedef __attribute__((ext_vector_type(2))) float v2f;
typedef __attribute__((ext_vector_type(8))) float v8f;

#define DFEAT 128
#define LDS_STRIDE 132   // 132 mod 64 banks = 4 -> conflict-free row reads

enum { EP_PLAIN = 0, EP_MASK = 1, EP_BIAS = 2 };

// ---------------------------------------------------------------------------
// C[nrows,128] = A[nrows,128] @ W[128,128], 16x16 tile per wave, K-step 4.
// mode EP_MASK : rows with mask==0 are replaced by aux (dec_token[128])
// mode EP_BIAS : aux (bias[128]) added to every row
// ---------------------------------------------------------------------------
__global__ __launch_bounds__(256) void gemm128_wmma_kernel(
    const float* __restrict__ A, const float* __restrict__ W,
    float* __restrict__ C, int nrows, int mode,
    const long long* __restrict__ mask, const float* __restrict__ aux)
{
    __shared__ float As[16 * LDS_STRIDE];
    const int tid  = threadIdx.x;
    const int row0 = blockIdx.x * 16;

    // Cooperatively stage the 16x128 A tile into LDS (padded stride).
    for (int f = tid; f < 512; f += 256) {          // 512 x float4 = 2048 floats
        const int r  = f >> 5;                      // tile row 0..15
        const int c4 = f & 31;                      // float4 column 0..31
        float4 v = make_float4(0.f, 0.f, 0.f, 0.f);
        const int row = row0 + r;
        if (row < nrows) v = ((const float4*)(A + (size_t)row * DFEAT))[c4];
        ((float4*)&As[r * LDS_STRIDE])[c4] = v;     // 528B row pitch: 16B aligned
    }
    __syncthreads();

    const int wave = tid >> 5;          // 8 waves: one 16-col tile each
    const int lane = tid & 31;
    const int half = lane >> 4;         // 0/1
    const int lm   = lane & 15;
    const int col  = wave * 16 + lm;

    // A 16x4 layout: lane(l) reg(r) -> M=l%16, K=2*(l/16)+r
    // B 4x16 layout: lane(l) reg(r) -> K=2*(l/16)+r, N=l%16
    v8f acc = {};
    for (int k = 0; k < DFEAT; k += 4) {
        const int kk = k + 2 * half;
        v2f a, b;
        float2 av = *(const float2*)&As[lm * LDS_STRIDE + kk];
        a.x = av.x;
        a.y = av.y;
        b.x = W[(size_t)kk * DFEAT + col];
        b.y = W[(size_t)(kk + 1) * DFEAT + col];
        acc = __builtin_amdgcn_wmma_f32_16x16x4_f32(
            false, a, false, b, (short)0, acc, false, false);
    }

    // D layout: reg r, lane l -> row = r + 8*(l/16), col = l%16 (+ tile bases)
    const float bias = (mode == EP_BIAS) ? aux[col] : 0.f;
    #pragma unroll
    for (int r = 0; r < 8; ++r) {
        const int row = row0 + r + 8 * half;
        if (row < nrows) {
            float v = acc[r] + bias;
            if (mode == EP_MASK && mask[row] == 0) v = aux[col];
            C[(size_t)row * DFEAT + col] = v;
        }
    }
}

// ---------------------------------------------------------------------------
// Degree / normalization (computed once, reused by both GCN layers).
// ---------------------------------------------------------------------------
__global__ void init_deg_kernel(unsigned* __restrict__ deg, int n) {
    int i = blockIdx.x * 256 + threadIdx.x;
    if (i < n) deg[i] = 1u;                 // self-loop
}
__global__ void accum_deg_kernel(const long long* __restrict__ ei, int E,
                                 unsigned* __restrict__ deg) {
    int e = blockIdx.x * 256 + threadIdx.x;
    if (e < E) atomicAdd(&deg[(int)ei[(size_t)E + e]], 1u);   // dst side
}
__global__ void dinv_kernel(const unsigned* __restrict__ deg,
                            float* __restrict__ dinv, int n) {
    int i = blockIdx.x * 256 + threadIdx.x;
    if (i < n) dinv[i] = rsqrtf((float)deg[i]);   // deg >= 1 always
}

// agg[n,128] = bias[128] broadcast (so scatter's "+b" is free)
__global__ void fill_bias_kernel(float* __restrict__ agg,
                                 const float* __restrict__ b, long long total) {
    long long i = (long long)blockIdx.x * 256 + threadIdx.x;
    if (i < total) agg[i] = b[i & (DFEAT - 1)];
}

// ---------------------------------------------------------------------------
// One wave32 per edge (E real edges + N self loops).
// Gather 128 floats of t[src] (float4/lane), scale by dinv[s]*dinv[d],
// scatter with global_atomic_add_f32 into agg[dst].
// ---------------------------------------------------------------------------
__global__ __launch_bounds__(256) void scatter_edges_kernel(
    const long long* __restrict__ ei, int E, int N,
    const float* __restrict__ dinv, const float* __restrict__ t,
    float* __restrict__ agg)
{
    long long e = (long long)blockIdx.x * 8 + (threadIdx.x >> 5);
    const int lane = threadIdx.x & 31;
    if (e >= (long long)E + N) return;

    int s, d;
    float nm;
    if (e < E) {
        s = (int)ei[e];
        d = (int)ei[(size_t)E + e];
        nm = dinv[s] * dinv[d];
    } else {
        s = d = (int)(e - E);
        const float di = dinv[s];
        nm = di * di;
    }
    const float4 v = ((const float4*)(t + (size_t)s * DFEAT))[lane];
    float* out = agg + (size_t)d * DFEAT + lane * 4;
    atomicAdd(out + 0, v.x * nm);
    atomicAdd(out + 1, v.y * nm);
    atomicAdd(out + 2, v.z * nm);
    atomicAdd(out + 3, v.w * nm);
}

// ---------------------------------------------------------------------------
// Fused LayerNorm + PReLU: one wave per node, 4 features/lane,
// wave32 butterfly reductions for mean and variance.
// ---------------------------------------------------------------------------
__global__ __launch_bounds__(256) void ln_prelu_kernel(
    const float* __restrict__ agg, const float* __restrict__ g,
    const float* __restrict__ be, const float* __restrict__ pa,
    float* __restrict__ out, int N)
{
    const int n = blockIdx.x * 8 + (threadIdx.x >> 5);
    if (n >= N) return;
    const int lane = threadIdx.x & 31;

    const float4 v = ((const float4*)(agg + (size_t)n * DFEAT))[lane];

    float s = v.x + v.y + v.z + v.w;
    #pragma unroll
    for (int m = 16; m >= 1; m >>= 1) s += __shfl_xor(s, m, 32);
    const float mu = s * (1.0f / DFEAT);

    const float dx = v.x - mu, dy = v.y - mu, dz = v.z - mu, dw = v.w - mu;
    float q = dx * dx + dy * dy + dz * dz + dw * dw;
    #pragma unroll
    for (int m = 16; m >= 1; m >>= 1) q += __shfl_xor(q, m, 32);
    const float rinv = rsqrtf(q * (1.0f / DFEAT) + 1e-5f);

    const float a = pa[0];
    const int c = lane * 4;
    float4 r;
    float y;
    y = dx * rinv * g[c + 0] + be[c + 0]; r.x = (y >= 0.f) ? y : a * y;
    y = dy * rinv * g[c + 1] + be[c + 1]; r.y = (y >= 0.f) ? y : a * y;
    y = dz * rinv * g[c + 2] + be[c + 2]; r.z = (y >= 0.f) ? y : a * y;
    y = dw * rinv * g[c + 3] + be[c + 3]; r.w = (y >= 0.f) ? y : a * y;
    ((float4*)(out + (size_t)n * DFEAT))[lane] = r;
}

// ---------------------------------------------------------------------------
extern "C" void kernel_launch(void* const* d_in, const int* in_sizes, int n_in,
                              void* d_out, int out_size, void* d_ws, size_t ws_size,
                              hipStream_t stream)
{
    const float*     x      = (const float*)d_in[0];
    const long long* ei     = (const long long*)d_in[1];
    const long long* mask   = (const long long*)d_in[2];
    const float*     W_enc  = (const float*)d_in[3];
    const float*     dtok   = (const float*)d_in[4];
    const float*     W1     = (const float*)d_in[5];
    const float*     b1     = (const float*)d_in[6];
    const float*     g1     = (const float*)d_in[7];
    const float*     be1    = (const float*)d_in[8];
    const float*     pa     = (const float*)d_in[9];
    const float*     W2     = (const float*)d_in[10];
    const float*     b2     = (const float*)d_in[11];
    const float*     g2     = (const float*)d_in[12];
    const float*     be2    = (const float*)d_in[13];
    const float*     Wo     = (const float*)d_in[14];
    const float*     bo     = (const float*)d_in[15];
    float*           out    = (float*)d_out;

    const int N = in_sizes[2];          // mask_vector has N elements
    const int E = in_sizes[1] / 2;      // edge_index is [2, E]
    const long long ND = (long long)N * DFEAT;

    // Workspace layout (256B aligned): deg | dinv | h | t | agg
    auto align256 = [](size_t v) { return (v + 255) & ~(size_t)255; };
    char* ws = (char*)d_ws;
    size_t off = 0;
    unsigned* deg  = (unsigned*)(ws + off); off = align256(off + (size_t)N * 4);
    float*    dinv = (float*)   (ws + off); off = align256(off + (size_t)N * 4);
    float*    h    = (float*)   (ws + off); off = align256(off + (size_t)ND * 4);
    float*    t    = (float*)   (ws + off); off = align256(off + (size_t)ND * 4);
    float*    agg  = (float*)   (ws + off); off = align256(off + (size_t)ND * 4);
    (void)ws_size; (void)n_in; (void)out_size;

    const int blkN   = (N + 255) / 256;
    const int blkE   = (E + 255) / 256;
    const int blkG   = (N + 15) / 16;                         // GEMM row tiles
    const int blkFill = (int)((ND + 255) / 256);
    const int blkScat = (int)(((long long)E + N + 7) / 8);    // 8 waves/block
    const int blkLN  = (N + 7) / 8;

    // --- degrees / symmetric normalization (shared by both layers) ---
    init_deg_kernel<<<blkN, 256, 0, stream>>>(deg, N);
    accum_deg_kernel<<<blkE, 256, 0, stream>>>(ei, E, deg);
    dinv_kernel<<<blkN, 256, 0, stream>>>(deg, dinv, N);

    // --- encoder GEMM + decoder-token mask injection ---
    gemm128_wmma_kernel<<<blkG, 256, 0, stream>>>(x, W_enc, h, N, EP_MASK, mask, dtok);

    // --- GCN layer 1 ---
    fill_bias_kernel<<<blkFill, 256, 0, stream>>>(agg, b1, ND);
    gemm128_wmma_kernel<<<blkG, 256, 0, stream>>>(h, W1, t, N, EP_PLAIN, nullptr, nullptr);
    scatter_edges_kernel<<<blkScat, 256, 0, stream>>>(ei, E, N, dinv, t, agg);
    ln_prelu_kernel<<<blkLN, 256, 0, stream>>>(agg, g1, be1, pa, h, N);

    // --- GCN layer 2 ---
    fill_bias_kernel<<<blkFill, 256, 0, stream>>>(agg, b2, ND);
    gemm128_wmma_kernel<<<blkG, 256, 0, stream>>>(h, W2, t, N, EP_PLAIN, nullptr, nullptr);
    scatter_edges_kernel<<<blkScat, 256, 0, stream>>>(ei, E, N, dinv, t, agg);
    ln_prelu_kernel<<<blkLN, 256, 0, stream>>>(agg, g2, be2, pa, h, N);

    // --- output projection ---
    gemm128_wmma_kernel<<<blkG, 256, 0, stream>>>(h, Wo, out, N, EP_BIAS, nullptr, bo);
}